// Always_6236292514511
// MI455X (gfx1250) — compile-verified
//
#include <hip/hip_runtime.h>
#include <stdint.h>

#define TAU_F     0.01f
#define INV_TAU   100.0f
#define TCOLS     16384
#define TILE      512                 // outputs per block
#define NCHUNK    (TILE / 16 + 4)     // 36 chunks of 16 cover tile + 63 halo
#define NLOAD     (NCHUNK * 16)       // 576 input elements staged per block
#define THREADS   256                 // 8 wave32 waves

typedef unsigned int v4u __attribute__((ext_vector_type(4)));
typedef int          v8i __attribute__((ext_vector_type(8)));
typedef int          v4i __attribute__((ext_vector_type(4)));

#if defined(__has_builtin)
# if __has_builtin(__builtin_amdgcn_tensor_load_to_lds)
#  define HAVE_TDM 1
# endif
#endif
#ifndef HAVE_TDM
# define HAVE_TDM 0
#endif

__device__ __forceinline__ void wait_tensorcnt0() {
#if defined(__has_builtin) && __has_builtin(__builtin_amdgcn_s_wait_tensorcnt)
  __builtin_amdgcn_s_wait_tensorcnt(0);
#else
  asm volatile("s_wait_tensorcnt 0x0" ::: "memory");
#endif
}

__global__ __launch_bounds__(THREADS)
void softmin_window_kernel(const float* __restrict__ sig,
                           float* __restrict__ out) {
  __shared__ float w_sh[NLOAD];           // staged signal tile (+inf padded)
  __shared__ float pmin_sh[NCHUNK * 16];  // running prefix min within chunk
  __shared__ float psum_sh[NCHUNK * 16];  // prefix sum of exp, rescaled by pmin
  __shared__ float smin_sh[NCHUNK * 16];  // running suffix min within chunk
  __shared__ float ssum_sh[NCHUNK * 16];  // suffix sum of exp, rescaled by smin
  __shared__ float mu_sh[NCHUNK];         // chunk min
  __shared__ float tot_sh[NCHUNK];        // chunk exp-total, rescaled by mu

  const int tid = threadIdx.x;
  const int row = blockIdx.y;
  const int t0  = blockIdx.x * TILE;      // first output column of this tile
  const float* grow = sig + (size_t)row * TCOLS + t0;

  // ---- Stage 1: async-stage 576 floats into LDS via the Tensor Data Mover ----
#if HAVE_TDM
  if (tid < 32) {   // one wave issues the TDM op (EXEC-independent, once/wave)
    const uint64_t ga   = (uint64_t)(uintptr_t)grow;
    const uint32_t ldsa = (uint32_t)(uintptr_t)w_sh;   // low 32 bits = LDS byte offset
    const uint32_t valid = (uint32_t)(TCOLS - t0);     // OOB-of-tensor reads return 0

    v4u g0;
    g0.x = 1u;                                          // count=1, user descriptor
    g0.y = ldsa;                                        // lds_addr [63:32]
    g0.z = (uint32_t)(ga & 0xffffffffu);                // global_addr lo
    g0.w = (uint32_t)((ga >> 32) & 0x01ffffffu)         // global_addr [56:32]
         | (2u << 30);                                  // type=2 ("image")

    v8i g1 = {0, 0, 0, 0, 0, 0, 0, 0};
    g1[0] = (int)(2u << 16);                            // data_size=2 -> 4 bytes
    g1[1] = (int)((valid & 0xffffu) << 16);             // tensor_dim0[15:0]
    g1[2] = (int)(((valid >> 16) & 0xffffu)             // tensor_dim0[31:16]
                  | (1u << 16));                        // tensor_dim1 = 1
    g1[3] = (int)((unsigned)NLOAD << 16);               // tile_dim0 = 576
    g1[4] = 1;                                          // tile_dim1=1, tile_dim2=0
    g1[5] = TCOLS;                                      // tensor_dim0_stride lo
    // g1[6], g1[7]: stride hi / dim1 stride = 0

    v4i gz4 = {0, 0, 0, 0};                             // groups 2/3 unused (2D)
    v8i gz8 = {0, 0, 0, 0, 0, 0, 0, 0};                 // extra group (clang-23 form)
    // clang-23 / therock toolchain: 6-arg form (v4u, v8i, v4i, v4i, v8i, i32 cpol)
    __builtin_amdgcn_tensor_load_to_lds(g0, g1, gz4, gz4, gz8, 0);
    wait_tensorcnt0();
  }
  __syncthreads();
  // Patch padding: TDM zero-fills past tensor end, reference wants +inf.
  for (int i = tid; i < NLOAD; i += THREADS)
    if (t0 + i >= TCOLS) w_sh[i] = INFINITY;
#else
  for (int i = tid; i < NLOAD; i += THREADS)
    w_sh[i] = (t0 + i < TCOLS) ? grow[i] : INFINITY;
#endif
  __syncthreads();

  // ---- Stage 2: per-chunk rescaled scans (1 exp per element per direction) ----
  // lanes [0,36): forward scan of chunk c; lanes [36,72): backward scan of c-36.
  if (tid < NCHUNK) {
    const int base = tid * 16;
    float m = INFINITY, p = 0.0f;
    #pragma unroll
    for (int i = 0; i < 16; ++i) {
      const float x = w_sh[base + i];
      if (x < m) {                      // rescale old sum to new reference min
        p = p * __expf((x - m) * INV_TAU) + 1.0f;   // exp(arg<=0): safe
        m = x;
      } else if (x < INFINITY) {        // skip inf-inf (NaN) on padded chunks
        p += __expf((m - x) * INV_TAU);
      }
      pmin_sh[base + i] = m;
      psum_sh[base + i] = p;
    }
    mu_sh[tid]  = m;
    tot_sh[tid] = p;
  } else if (tid < 2 * NCHUNK) {
    const int base = (tid - NCHUNK) * 16;
    float m = INFINITY, p = 0.0f;
    #pragma unroll
    for (int i = 15; i >= 0; --i) {
      const float x = w_sh[base + i];
      if (x < m) {
        p = p * __expf((x - m) * INV_TAU) + 1.0f;
        m = x;
      } else if (x < INFINITY) {
        p += __expf((m - x) * INV_TAU);
      }
      smin_sh[base + i] = m;
      ssum_sh[base + i] = p;
    }
  }
  __syncthreads();

  // ---- Stage 3: combine 5 segments per output (5 exp + 1 log) ----
  #pragma unroll
  for (int o = 0; o < TILE / THREADS; ++o) {
    const int t  = tid + o * THREADS;   // tile-local output index
    const int q  = t & 15;
    const int c0 = t >> 4;

    const float hm = smin_sh[c0 * 16 + q];      // head: chunk c0, elems q..15
    const float hs = ssum_sh[c0 * 16 + q];
    const float m1 = mu_sh[c0 + 1], s1 = tot_sh[c0 + 1];
    const float m2 = mu_sh[c0 + 2], s2 = tot_sh[c0 + 2];
    const float m3 = mu_sh[c0 + 3], s3 = tot_sh[c0 + 3];
    float tm = INFINITY, ts = 0.0f;             // tail: chunk c0+4, elems 0..q-1
    if (q > 0) {
      tm = pmin_sh[(c0 + 4) * 16 + q - 1];
      ts = psum_sh[(c0 + 4) * 16 + q - 1];
    }

    // window min (head contains x[t] itself, so m is always finite)
    const float m = fminf(fminf(fminf(hm, tm), fminf(m1, m2)), m3);

    // all scale exponents are <= 0; inf-min segments have sum==0 -> 0*0 = 0
    float S = hs * __expf((m - hm) * INV_TAU);
    S += s1 * __expf((m - m1) * INV_TAU);
    S += s2 * __expf((m - m2) * INV_TAU);
    S += s3 * __expf((m - m3) * INV_TAU);
    S += ts * __expf((m - tm) * INV_TAU);

    out[(size_t)row * TCOLS + t0 + t] = m - TAU_F * __logf(S);
  }
}

extern "C" void kernel_launch(void* const* d_in, const int* in_sizes, int n_in,
                              void* d_out, int out_size, void* d_ws, size_t ws_size,
                              hipStream_t stream) {
  const float* sig = (const float*)d_in[0];
  float* out = (float*)d_out;
  const int rows = in_sizes[0] / TCOLS;   // 64
  dim3 grid(TCOLS / TILE, rows);          // 32 x 64 blocks
  softmin_window_kernel<<<grid, THREADS, 0, stream>>>(sig, out);
}